// MaxPool_9311489098145
// MI455X (gfx1250) — compile-verified
//
#include <hip/hip_runtime.h>
#include <stdint.h>

// Reference: x [32,64,224,224] f32 -> out [32,64,112,112] f32
// Quirky unfold/reshape maxpool reduces to:
//   out_flat[p*12544 + (2*ki+kj)*3136 + oy*28 + mx]
//     = max_{j=0..3} x[p, ki+2*oy, 8*mx + kj + 2*j]
// i.e. each 8 consecutive input floats (a "task") -> 2 outputs:
//   even-lane max -> kj=0 quarter, odd-lane max -> kj=1 quarter.
// Pure streaming op: ~514 MB traffic, HBM-bound (~22 us at 23.3 TB/s).
// Use CDNA5 async global->LDS copies (double buffered) + NT stores.

constexpr int kBlocks        = 1024;
constexpr int kThreads       = 256;
constexpr int kWavesPerBlock = kThreads / 32;                 // 8 wave32
constexpr int kNWaves        = kBlocks * kWavesPerBlock;      // 8192
constexpr int kTasks         = 32 * 64 * 224 * 28;            // 12,845,056
constexpr int kChunks        = kTasks / 32;                   // 401,408 (1KB each)
constexpr int kIters         = kChunks / kNWaves;             // 49 exactly
static_assert(kChunks % kNWaves == 0, "no tail by construction");

__global__ __launch_bounds__(kThreads) void unfold_maxpool_kernel(
    const float* __restrict__ x, float* __restrict__ out) {
  // 8 waves * 2 buffers * 1KB = 16KB LDS per block
  __shared__ alignas(16) float smem[4096];

  const int lane = threadIdx.x & 31;
  const int wib  = threadIdx.x >> 5;
  const int gw   = blockIdx.x * kWavesPerBlock + wib;   // global wave id

  // Generic pointer to __shared__ has the LDS byte offset in its low 32 bits.
  const uint32_t lds_wave = (uint32_t)(uintptr_t)smem + (uint32_t)wib * 2048u;

  // Prologue: async-copy chunk for iteration 0 into buffer 0.
  {
    const uint32_t ldst = lds_wave + (uint32_t)lane * 16u;
    const uint32_t goff = (uint32_t)gw * 1024u + (uint32_t)lane * 16u;
    asm volatile(
        "global_load_async_to_lds_b128 %0, %1, %2\n\t"
        "global_load_async_to_lds_b128 %0, %1, %2 offset:512"
        :: "v"(ldst), "v"(goff), "s"(x) : "memory");
  }

  for (int k = 0; k < kIters; ++k) {
    const int buf = k & 1;

    if (k + 1 < kIters) {
      // Issue next chunk into the other buffer, then wait for current chunk.
      const uint32_t ldst = lds_wave + (uint32_t)((k + 1) & 1) * 1024u
                          + (uint32_t)lane * 16u;
      const uint32_t goff = (uint32_t)(gw + (k + 1) * kNWaves) * 1024u
                          + (uint32_t)lane * 16u;
      asm volatile(
          "global_load_async_to_lds_b128 %0, %1, %2\n\t"
          "global_load_async_to_lds_b128 %0, %1, %2 offset:512"
          :: "v"(ldst), "v"(goff), "s"(x) : "memory");
      // Async loads complete in order: <=2 outstanding => current chunk landed.
      asm volatile("s_wait_asynccnt 0x2" ::: "memory");
    } else {
      asm volatile("s_wait_asynccnt 0x0" ::: "memory");
    }

    // This lane's 8 floats (32B) sit linearly in the wave's buffer.
    const float4* lp = reinterpret_cast<const float4*>(
        &smem[wib * 512 + buf * 256 + lane * 8]);
    const float4 a = lp[0];
    const float4 b = lp[1];
    const float o_even = fmaxf(fmaxf(a.x, a.z), fmaxf(b.x, b.z));  // kj = 0
    const float o_odd  = fmaxf(fmaxf(a.y, a.w), fmaxf(b.y, b.w));  // kj = 1

    // task -> (plane p, input row r, mx)
    const int task = (gw + k * kNWaves) * 32 + lane;
    const int mx = task % 28;
    const int t2 = task / 28;
    const int r  = t2 % 224;
    const int p  = t2 / 224;
    // out_flat = p*12544 + (2*ki+kj)*3136 + oy*28 + mx
    const int ob = (p * 4 + 2 * (r & 1)) * 3136 + (r >> 1) * 28 + mx;

    // Output is write-once: bypass cache retention with NT stores.
    __builtin_nontemporal_store(o_even, out + ob);
    __builtin_nontemporal_store(o_odd,  out + ob + 3136);
  }
}

extern "C" void kernel_launch(void* const* d_in, const int* in_sizes, int n_in,
                              void* d_out, int out_size, void* d_ws, size_t ws_size,
                              hipStream_t stream) {
  const float* x = (const float*)d_in[0];
  float* out = (float*)d_out;
  (void)in_sizes; (void)n_in; (void)out_size; (void)d_ws; (void)ws_size;
  unfold_maxpool_kernel<<<kBlocks, kThreads, 0, stream>>>(x, out);
}